// CliffordLinearEquivariant_13950053777609
// MI455X (gfx1250) — compile-verified
//
#include <hip/hip_runtime.h>

typedef __attribute__((ext_vector_type(2))) float v2f;
typedef __attribute__((ext_vector_type(8))) float v8f;

#define DIM  8
#define NI   16
#define NO   16
#define KW   2
#define KDIM 128   /* I*DIM : reduction dim  */
#define MDIM 128   /* O*DIM : output columns */

// ---------------------------------------------------------------------------
// Cayley reordering sign for Cl(3,0), Euclidean metric (all +1, so the
// metric contributes nothing beyond the transposition count).
// ---------------------------------------------------------------------------
__device__ __forceinline__ float reorder_sign(int a, int b) {
    int s = 0;
    int aa = a >> 1;
    while (aa) { s += __popc(aa & b); aa >>= 1; }
    return (s & 1) ? -1.0f : 1.0f;
}

// ---------------------------------------------------------------------------
// Build T2[k][m] (128 x 128), k = i*8+q, m = o*8+r:
//   T2 = sum_{kk,p,s} sign(p,q)*sign(p^q,s) * wl[kk,o,i,p] * wr[kk,o,i,s]
// using C[p,q,m] != 0  <=>  m == p^q  (so r = p^q^s).
// One thread per (o,i) pair; single 256-thread block. Tiny one-off cost.
// ---------------------------------------------------------------------------
__global__ void build_T2(const float* __restrict__ wl,
                         const float* __restrict__ wr,
                         float* __restrict__ T2) {
    int oi = threadIdx.x;          // 0..255
    int o  = oi >> 4;
    int i  = oi & 15;

    float acc[DIM][DIM];
    for (int q = 0; q < DIM; ++q)
        for (int r = 0; r < DIM; ++r) acc[q][r] = 0.0f;

    for (int kk = 0; kk < KW; ++kk) {
        const float* a = wl + (((kk * NO + o) * NI + i) * DIM);
        const float* b = wr + (((kk * NO + o) * NI + i) * DIM);
        float av[DIM], bv[DIM];
        for (int p = 0; p < DIM; ++p) { av[p] = a[p]; bv[p] = b[p]; }

        for (int p = 0; p < DIM; ++p) {
            for (int q = 0; q < DIM; ++q) {
                int   m  = p ^ q;
                float s1 = reorder_sign(p, q) * av[p];
                for (int s = 0; s < DIM; ++s) {
                    int r = m ^ s;
                    acc[q][r] += s1 * reorder_sign(m, s) * bv[s];
                }
            }
        }
    }

    for (int q = 0; q < DIM; ++q)
        for (int r = 0; r < DIM; ++r)
            T2[(i * DIM + q) * MDIM + (o * DIM + r)] = acc[q][r];
}

// ---------------------------------------------------------------------------
// Streaming GEMM: out[n, m] = sum_k x[n,k] * T2[k,m] + bias[m]
//   N = 262144 rows, K = M = 128.
// Block = 256 threads = 8 wave32s; each wave computes a 16x128 output tile
// with v_wmma_f32_16x16x4_f32 (fp32 in / fp32 accum keeps full precision;
// the problem is HBM-bound at ~32 FLOP/byte so fp32 WMMA is already free).
//
// T2 is staged into LDS in B-fragment order so each lane's B operand is a
// contiguous ds_load_b64:
//   k = kk*4 + h*2 + j  ->  ldsB[(kk*2 + h)*256 + m*2 + j]
// matching the ISA B layout (VGPR index <-> K row, lanes 0-15/16-31 <-> N,
// K-half select by lane>>4).
// ---------------------------------------------------------------------------
__global__ __launch_bounds__(256) void clifford_gemm(
        const float* __restrict__ x,
        const float* __restrict__ T2,
        const float* __restrict__ bias,
        float* __restrict__ out) {
    __shared__ float ldsB[KDIM * MDIM];   // 64 KB of the 320 KB WGP LDS

    // ---- stage T2 into LDS, pre-swizzled into WMMA B-fragment layout ----
    for (int g = threadIdx.x; g < KDIM * MDIM; g += 256) {
        int k = g >> 7;
        int m = g & 127;
        int slot = (((k >> 2) * 2 + ((k >> 1) & 1)) * 256) + m * 2 + (k & 1);
        ldsB[slot] = T2[g];
    }
    __syncthreads();

    const int wave = threadIdx.x >> 5;   // 0..7 (wave32)
    const int lane = threadIdx.x & 31;
    const int ln   = lane & 15;          // N column / A row within half
    const int h    = lane >> 4;          // K-half select

    const int n0 = blockIdx.x * 128 + wave * 16;   // 16 rows for this wave

    // ---- init accumulators with bias (bias depends only on column N) ----
    v8f acc[8];
#pragma unroll
    for (int mt = 0; mt < 8; ++mt) {
        float bv = bias[mt * 16 + ln];
        v8f t = {bv, bv, bv, bv, bv, bv, bv, bv};
        acc[mt] = t;
    }

    // ---- K loop: 32 steps of 4; A = x fragment, B from LDS ----
    // A layout (32-bit 16x4): lane l row = l&15; VGPR0/1 = K (l>>4)*2, +1.
    const float* xp = x + (size_t)(n0 + ln) * KDIM + h * 2;
#pragma unroll 4
    for (int kk = 0; kk < KDIM / 4; ++kk) {
        v2f a = *(const v2f*)(xp + kk * 4);
        int bbase = (kk * 2 + h) * 256 + ln * 2;
#pragma unroll
        for (int mt = 0; mt < 8; ++mt) {
            v2f b = *(const v2f*)(&ldsB[bbase + mt * 32]);
            acc[mt] = __builtin_amdgcn_wmma_f32_16x16x4_f32(
                /*neg_a=*/false, a, /*neg_b=*/false, b,
                /*c_mod=*/(short)0, acc[mt],
                /*reuse_a=*/false, /*reuse_b=*/false);
        }
    }

    // ---- store: C/D layout -> VGPR v is row v + h*8, column = mt*16 + ln.
    // Lanes 0-15 and 16-31 each write dense 64B segments per instruction.
    size_t nbase = (size_t)n0 * MDIM;
#pragma unroll
    for (int mt = 0; mt < 8; ++mt) {
#pragma unroll
        for (int v = 0; v < 8; ++v) {
            out[nbase + (size_t)(v + h * 8) * MDIM + mt * 16 + ln] = acc[mt][v];
        }
    }
}

// ---------------------------------------------------------------------------
// Launch: inputs are x, weight_left, weight_right, bias (all fp32).
// d_ws holds T2 (64 KB).
// ---------------------------------------------------------------------------
extern "C" void kernel_launch(void* const* d_in, const int* in_sizes, int n_in,
                              void* d_out, int out_size, void* d_ws, size_t ws_size,
                              hipStream_t stream) {
    const float* x  = (const float*)d_in[0];
    const float* wl = (const float*)d_in[1];
    const float* wr = (const float*)d_in[2];
    const float* bs = (const float*)d_in[3];
    float* out = (float*)d_out;
    float* T2  = (float*)d_ws;

    build_T2<<<1, 256, 0, stream>>>(wl, wr, T2);

    int rows   = in_sizes[0] / KDIM;   // B*S = 262144
    int blocks = rows / 128;           // 2048 blocks of 8 waves
    clifford_gemm<<<blocks, 256, 0, stream>>>(x, T2, bs, out);
}